// EFConv_21921513079142
// MI455X (gfx1250) — compile-verified
//
#include <hip/hip_runtime.h>

// EFConv fused kernels for gfx1250 (CDNA5, wave32, WMMA f32 16x16x4)

typedef __attribute__((ext_vector_type(2))) float v2f;
typedef __attribute__((ext_vector_type(8))) float v8f;

#define NB 16
#define NA 128
#define HD 256
#define EF 30
#define KP 32          // padded K (30 -> 32) for filter GEMM
#define K2 (2 * HD)    // 512, K of node MLP
#define KC 32          // K chunk for node MLP
#define ROWS 64        // rows per block in node MLP

__device__ __forceinline__ float selu_f(float x) {
    const float scale = 1.0507009873554805f;
    const float alpha = 1.6732632423543772f;
    return x > 0.0f ? scale * x : scale * alpha * (__expf(x) - 1.0f);
}

__device__ __forceinline__ v8f wmma_f32(v2f a, v2f b, v8f c) {
    // 8 args: (neg_a, A, neg_b, B, c_mod, C, reuse_a, reuse_b)
    return __builtin_amdgcn_wmma_f32_16x16x4_f32(false, a, false, b, (short)0, c,
                                                 false, false);
}

// ---------------------------------------------------------------------------
// Kernel 1: fused  e_f = selu(e @ Wf^T + bf) * edge_mask ;
//                  h1[b,i,f] = sum_j e_f[b,i,j,f] * h[b,j,f]
// One block per (b,i): 128(j) x 256(f) GEMM with K=32 (padded), then the
// j-reduction done straight out of the WMMA accumulator layout.
// Writes h1 into `h1out` (= d_out, consumed by kernel 2).
// ---------------------------------------------------------------------------
__global__ __launch_bounds__(256)
void ef_filter_reduce(const float* __restrict__ h,
                      const float* __restrict__ e,
                      const float* __restrict__ edge_mask,
                      const float* __restrict__ Wf,
                      const float* __restrict__ bf,
                      float* __restrict__ h1out) {
    __shared__ float sh_wf[KP * HD];  // [k][f], zero padded k>=30  (32 KB)
    __shared__ float sh_e[NA * KP];   // [j][k], zero padded        (16 KB)

    const int tid = threadIdx.x;
    const int bi  = blockIdx.x;
    const int b   = bi / NA;
    const int i   = bi % NA;

    // Stage Wf transposed into B layout [k][f]; thread tid owns column f=tid.
    {
        const float* wrow = Wf + (size_t)tid * EF;
#pragma unroll
        for (int k = 0; k < KP; ++k)
            sh_wf[k * HD + tid] = (k < EF) ? wrow[k] : 0.0f;
    }
    // Stage e[b,i,:,:] (128 x 30) into [j][32] with zero pad.
    {
        const float* erow = e + ((size_t)(b * NA + i)) * NA * EF;
        const int j  = tid >> 1;
        const int kb = (tid & 1) * 16;
#pragma unroll
        for (int k = 0; k < 16; ++k) {
            const int kk = kb + k;
            sh_e[j * KP + kk] = (kk < EF) ? erow[j * EF + kk] : 0.0f;
        }
    }
    __syncthreads();

    const int lane  = tid & 31;
    const int wave  = tid >> 5;
    const int half  = lane >> 4;   // which 16-lane half
    const int l     = lane & 15;
    const int fbase = wave * 32;   // each wave owns 32 f columns (2 N tiles)

    const v8f vzero = {0.f, 0.f, 0.f, 0.f, 0.f, 0.f, 0.f, 0.f};
    v8f acc[8][2];                 // 8 M-tiles (j) x 2 N-tiles (f)
#pragma unroll
    for (int mt = 0; mt < 8; ++mt)
#pragma unroll
        for (int nt = 0; nt < 2; ++nt)
            acc[mt][nt] = vzero;

#pragma unroll
    for (int kk = 0; kk < KP / 4; ++kk) {
        const int k0 = kk * 4 + 2 * half;
#pragma unroll
        for (int mt = 0; mt < 8; ++mt) {
            const int j = mt * 16 + l;
            v2f a;
            a.x = sh_e[j * KP + k0];
            a.y = sh_e[j * KP + k0 + 1];
#pragma unroll
            for (int nt = 0; nt < 2; ++nt) {
                const int n = fbase + nt * 16 + l;
                v2f bb;
                bb.x = sh_wf[k0 * HD + n];
                bb.y = sh_wf[(k0 + 1) * HD + n];
                acc[mt][nt] = wmma_f32(a, bb, acc[mt][nt]);
            }
        }
    }

    // Reduce over j: lane (l,half) of tile (mt,nt) holds E[j, f] for
    // j = mt*16 + 8*half + r (r=0..7), f = fbase + nt*16 + l.
    const float* hB  = h + (size_t)b * NA * HD;
    const float* emB = edge_mask + ((size_t)(b * NA + i)) * NA;
#pragma unroll
    for (int nt = 0; nt < 2; ++nt) {
        const int f     = fbase + nt * 16 + l;
        const float bia = bf[f];
        float psum = 0.0f;
#pragma unroll
        for (int mt = 0; mt < 8; ++mt) {
#pragma unroll
            for (int r = 0; r < 8; ++r) {
                const int j  = mt * 16 + half * 8 + r;
                const float ef = selu_f(acc[mt][nt][r] + bia) * emB[j];
                psum += ef * hB[j * HD + f];
            }
        }
        const float tot = psum + __shfl_xor(psum, 16, 32);
        if (half == 0)
            h1out[(size_t)(b * NA + i) * HD + f] = tot;
    }
}

// ---------------------------------------------------------------------------
// Kernel 2: out = selu([h, h1] @ Ww^T + bw) * node_mask + h
// M=2048 rows, N=256, K=512. 64 rows x 256 cols per block, K streamed in
// 32-wide chunks through LDS. h1 is read from `out` (written by kernel 1)
// only for this block's own rows, then overwritten in the epilogue.
// ---------------------------------------------------------------------------
__global__ __launch_bounds__(256)
void ef_node_mlp(const float* __restrict__ h,
                 const float* __restrict__ node_mask,
                 const float* __restrict__ Ww,
                 const float* __restrict__ bw,
                 float* __restrict__ out /* holds h1 on entry */) {
    __shared__ float sh_z[ROWS * KC];  // 8 KB
    __shared__ float sh_w[KC * HD];    // 32 KB

    const int tid = threadIdx.x;
    const int r0  = blockIdx.x * ROWS;

    const int lane  = tid & 31;
    const int wave  = tid >> 5;
    const int half  = lane >> 4;
    const int l     = lane & 15;
    const int nbase = wave * 32;

    const v8f vzero = {0.f, 0.f, 0.f, 0.f, 0.f, 0.f, 0.f, 0.f};
    v8f acc[4][2];
#pragma unroll
    for (int mt = 0; mt < 4; ++mt)
#pragma unroll
        for (int nt = 0; nt < 2; ++nt)
            acc[mt][nt] = vzero;

    for (int kc = 0; kc < K2 / KC; ++kc) {
        const int kg0 = kc * KC;
        // Stage z chunk: z[row][k] = (kg<HD) ? h[row][kg] : h1[row][kg-HD]
        {
            const int rr  = tid >> 2;        // 0..63
            const int kb  = (tid & 3) * 8;   // 0,8,16,24
            const int row = r0 + rr;
#pragma unroll
            for (int k = 0; k < 8; ++k) {
                const int kg = kg0 + kb + k;
                float v;
                if (kg < HD) v = h[(size_t)row * HD + kg];
                else         v = out[(size_t)row * HD + (kg - HD)];
                sh_z[rr * KC + kb + k] = v;
            }
        }
        // Stage Ww chunk as B layout [k][n]; thread tid owns n=tid.
        {
            const float* wn = Ww + (size_t)tid * K2 + kg0;
#pragma unroll
            for (int k = 0; k < KC; ++k)
                sh_w[k * HD + tid] = wn[k];
        }
        __syncthreads();

#pragma unroll
        for (int kk = 0; kk < KC / 4; ++kk) {
            const int k0 = kk * 4 + 2 * half;
#pragma unroll
            for (int mt = 0; mt < 4; ++mt) {
                const int rr = mt * 16 + l;
                v2f a;
                a.x = sh_z[rr * KC + k0];
                a.y = sh_z[rr * KC + k0 + 1];
#pragma unroll
                for (int nt = 0; nt < 2; ++nt) {
                    const int n = nbase + nt * 16 + l;
                    v2f bb;
                    bb.x = sh_w[k0 * HD + n];
                    bb.y = sh_w[(k0 + 1) * HD + n];
                    acc[mt][nt] = wmma_f32(a, bb, acc[mt][nt]);
                }
            }
        }
        __syncthreads();
    }

    // Epilogue: selu + bias, node mask, residual. Overwrites h1 rows owned
    // exclusively by this block (all h1 reads already done above).
#pragma unroll
    for (int mt = 0; mt < 4; ++mt) {
#pragma unroll
        for (int nt = 0; nt < 2; ++nt) {
            const int f = nbase + nt * 16 + l;
#pragma unroll
            for (int r = 0; r < 8; ++r) {
                const int row = r0 + mt * 16 + half * 8 + r;
                float v = selu_f(acc[mt][nt][r] + bw[f]) * node_mask[row];
                out[(size_t)row * HD + f] = v + h[(size_t)row * HD + f];
            }
        }
    }
}

extern "C" void kernel_launch(void* const* d_in, const int* in_sizes, int n_in,
                              void* d_out, int out_size, void* d_ws, size_t ws_size,
                              hipStream_t stream) {
    (void)in_sizes; (void)n_in; (void)out_size; (void)d_ws; (void)ws_size;
    const float* h         = (const float*)d_in[0];
    const float* e         = (const float*)d_in[1];
    const float* node_mask = (const float*)d_in[2];
    const float* edge_mask = (const float*)d_in[3];
    const float* Wf        = (const float*)d_in[4];
    const float* bf        = (const float*)d_in[5];
    const float* Ww        = (const float*)d_in[6];
    const float* bw        = (const float*)d_in[7];
    float* out = (float*)d_out;

    // Stage 1: fused filter-net GEMM + edge reduction -> h1 in d_out.
    ef_filter_reduce<<<NB * NA, 256, 0, stream>>>(h, e, edge_mask, Wf, bf, out);
    // Stage 2: gated node MLP + residual, in-place on d_out.
    ef_node_mlp<<<(NB * NA) / ROWS, 256, 0, stream>>>(h, node_mask, Ww, bw, out);
}